// SpikeFP32Sqrt_41085657153680
// MI455X (gfx1250) — compile-verified
//
#include <hip/hip_runtime.h>
#include <stdint.h>

// ---------------------------------------------------------------------------
// Bit-level FP32 sqrt emulation matching the JAX gate-network reference.
// Memory-bound problem: 32 MB total traffic -> ~1.4 us floor at 23.3 TB/s.
// CDNA5 paths used: wave32 ballot (bit transpose) + async global->LDS DMA.
// ---------------------------------------------------------------------------

#define WAVES_PER_BLOCK 8
#define TPB 256
#define ROWS_PER_TILE 8                       // 8 rows * 128 B = 1 KB per buffer
#define TILE_FLOATS (ROWS_PER_TILE * 32)      // 256 floats
#define TILE_BYTES (TILE_FLOATS * 4)          // 1024 bytes

__device__ __forceinline__ unsigned ballot32(bool p) {
#if __has_builtin(__builtin_amdgcn_ballot_w32)
  return __builtin_amdgcn_ballot_w32(p);
#else
  return (unsigned)__ballot(p);
#endif
}

// Async DMA: 16 bytes/lane, global (saddr base + vaddr byte offset) -> LDS.
__device__ __forceinline__ void async_load_b128(unsigned lds_byte_off,
                                                const void* sbase,
                                                unsigned gbyte_off) {
  asm volatile("global_load_async_to_lds_b128 %0, %1, %2"
               :
               : "v"(lds_byte_off), "v"(gbyte_off), "s"(sbase)
               : "memory");
}
__device__ __forceinline__ void wait_async_le2() {
  asm volatile("s_wait_asynccnt 2" ::: "memory");
}
__device__ __forceinline__ void wait_async_0() {
  asm volatile("s_wait_asynccnt 0" ::: "memory");
}

// Exact emulation of the reference network, including its quirks:
//   +-0 -> +0 ; +-Inf -> +Inf (inf mux applied last in the reference);
//   NaN or negative non-zero -> 0x7FC00000 ; subnormals get implicit 1.
__device__ __forceinline__ uint32_t ref_sqrt_bits(uint32_t u) {
  const uint32_t s = u >> 31;
  const uint32_t e = (u >> 23) & 0xFFu;
  const uint32_t m = u & 0x7FFFFFu;
  if (e == 0u && m == 0u) return 0u;                       // +-0 -> +0
  if (e == 255u) return m ? 0x7FC00000u : 0x7F800000u;     // NaN / +-Inf->+Inf
  if (s) return 0x7FC00000u;                               // negative -> qNaN
  const uint32_t mant24 = m | 0x800000u;
  const unsigned sh = 25u + ((~e) & 1u);                   // unbiased exp odd
  const unsigned long long a = (unsigned long long)mant24 << sh;  // < 2^51
  // floor(sqrt(a)): correctly-rounded f64 sqrt + +-1 fixup (branches uniform).
  uint32_t q = (uint32_t)sqrt((double)a);
  while ((unsigned long long)q * q > a) --q;
  while ((unsigned long long)(q + 1u) * (q + 1u) <= a) ++q;
  const unsigned long long rem = a - (unsigned long long)q * q;
  uint32_t mant23 = (q >> 1) & 0x7FFFFFu;                  // drop implicit 1
  const uint32_t rb = q & 1u;
  const uint32_t sticky = rem ? 1u : 0u;
  mant23 = (mant23 + (rb & (sticky | (mant23 & 1u)))) & 0x7FFFFFu;  // RNE
  const uint32_t ef = (e + 126u + (e & 1u)) >> 1;
  return (ef << 23) | mant23;
}

__global__ __launch_bounds__(TPB) void bitsqrt_kernel(
    const float* __restrict__ x, float* __restrict__ out, unsigned n_rows) {
  __shared__ float smem[WAVES_PER_BLOCK * 2 * TILE_FLOATS];  // 16 KB

  const unsigned lane = threadIdx.x & 31u;
  const unsigned wave = threadIdx.x >> 5;
  const unsigned wgid = blockIdx.x * WAVES_PER_BLOCK + wave;
  const unsigned nw = gridDim.x * WAVES_PER_BLOCK;
  const unsigned n_tiles = n_rows / ROWS_PER_TILE;

  // LDS byte offset of this wave's private double buffer (low 32 bits of the
  // shared-aperture flat address are the allocation-relative LDS offset).
  const unsigned ldsWaveBase =
      (unsigned)(uintptr_t)(void*)smem + wave * (2u * TILE_BYTES);
  const unsigned fWaveBase = wave * (2u * TILE_FLOATS);

  unsigned tile = wgid;
  unsigned pb = 0;
  if (tile < n_tiles) {  // prime buffer 0
    const unsigned o = lane * 16u;
    const unsigned g = tile * TILE_BYTES;
    async_load_b128(ldsWaveBase + o, x, g + o);
    async_load_b128(ldsWaveBase + o + 512u, x, g + o + 512u);
  }
  for (; tile < n_tiles; tile += nw) {
    const unsigned nxt = tile + nw;
    if (nxt < n_tiles) {  // prefetch next tile into the other buffer
      const unsigned o = lane * 16u;
      const unsigned g = nxt * TILE_BYTES;
      const unsigned lb = ldsWaveBase + (pb ^ 1u) * TILE_BYTES;
      async_load_b128(lb + o, x, g + o);
      async_load_b128(lb + o + 512u, x, g + o + 512u);
      wait_async_le2();  // current tile's 2 DMAs done; prefetch stays in flight
    } else {
      wait_async_0();
    }
    const float* b = smem + fWaveBase + pb * TILE_FLOATS;
    const size_t rowbase = (size_t)tile * ROWS_PER_TILE;
#pragma unroll
    for (int r = 0; r < ROWS_PER_TILE; ++r) {
      // lane l holds bit (31-l): ballot across wave32 IS the packed word.
      const float v = b[r * 32 + (31u - lane)];
      const unsigned mask = ballot32(v != 0.0f);
      const unsigned wbits = ref_sqrt_bits(mask);  // uniform across wave
      out[(rowbase + (unsigned)r) * 32 + lane] =
          (float)((wbits >> (31u - lane)) & 1u);
    }
    pb ^= 1u;
  }

  // Tail rows (n_rows not a multiple of ROWS_PER_TILE) via direct global path.
  for (size_t row = (size_t)n_tiles * ROWS_PER_TILE + wgid; row < n_rows;
       row += nw) {
    const float v = x[row * 32 + (31u - lane)];
    const unsigned mask = ballot32(v != 0.0f);
    const unsigned wbits = ref_sqrt_bits(mask);
    out[row * 32 + lane] = (float)((wbits >> (31u - lane)) & 1u);
  }
}

extern "C" void kernel_launch(void* const* d_in, const int* in_sizes, int n_in,
                              void* d_out, int out_size, void* d_ws,
                              size_t ws_size, hipStream_t stream) {
  (void)n_in; (void)out_size; (void)d_ws; (void)ws_size;
  const float* x = (const float*)d_in[0];
  float* out = (float*)d_out;
  const unsigned n_rows = (unsigned)(in_sizes[0] / 32);
  const unsigned n_tiles = n_rows / ROWS_PER_TILE;
  unsigned blocks = (n_tiles + WAVES_PER_BLOCK - 1) / WAVES_PER_BLOCK;
  if (blocks > 512u) blocks = 512u;   // 4096 waves -> ~4 pipelined tiles/wave
  if (blocks == 0u) blocks = 1u;
  bitsqrt_kernel<<<dim3(blocks), dim3(TPB), 0, stream>>>(x, out, n_rows);
}